// HVMBlock_1915555414835
// MI455X (gfx1250) — compile-verified
//
#include <hip/hip_runtime.h>

typedef float v2f __attribute__((ext_vector_type(2)));
typedef float v8f __attribute__((ext_vector_type(8)));

constexpr int B_ = 8, H_ = 56, W_ = 56, C_ = 96;
constexpr int D_ = 192, N_ = 16, R_ = 6, K_ = 4;
constexpr int L_  = H_ * W_;        // 3136
constexpr int NP_ = B_ * L_;        // 25088
constexpr int CD_ = R_ + 2 * N_;    // 38
constexpr int OO_ = 2 * D_;         // 384
constexpr int CH_ = 64;             // scan chunk length
constexpr int S_  = L_ / CH_;       // 49 chunks
constexpr int BK_ = B_ * K_;        // 32
constexpr float EPS_ = 1e-6f;

// ---------------- WMMA helper: native fp32 16x16x4 ----------------
__device__ __forceinline__ v8f wmma4(v2f a, v2f b, v8f c) {
  return __builtin_amdgcn_wmma_f32_16x16x4_f32(false, a, false, b, (short)0, c,
                                               false, false);
}

// scan-step t of direction k -> physical pixel index p = h*W + w
__device__ __forceinline__ int map_scan(int t, int k) {
  if (k >= 2) t = L_ - 1 - t;
  if (k & 1) { int hh = t % H_; int ww = t / H_; return hh * W_ + ww; }
  return t;
}

__device__ __forceinline__ float wave_sum(float v) {
#pragma unroll
  for (int off = 16; off > 0; off >>= 1) v += __shfl_xor(v, off, 32);
  return v;
}

// ---------------- 1) LayerNorm over C=96, one wave per pixel ----------------
__global__ void k_ln1(const float* __restrict__ x, const float* __restrict__ w,
                      const float* __restrict__ bb, float* __restrict__ xn) {
  int wave = (blockIdx.x * blockDim.x + threadIdx.x) >> 5;
  int lane = threadIdx.x & 31;
  if (wave >= NP_) return;
  const float* xp = x + (long)wave * C_;
  float v0 = xp[lane], v1 = xp[lane + 32], v2 = xp[lane + 64];
  float mean = wave_sum(v0 + v1 + v2) * (1.f / C_);
  float d0 = v0 - mean, d1 = v1 - mean, d2 = v2 - mean;
  float var = wave_sum(d0 * d0 + d1 * d1 + d2 * d2) * (1.f / C_);
  float rstd = rsqrtf(var + EPS_);
  float* op = xn + (long)wave * C_;
  op[lane]      = d0 * rstd * w[lane]      + bb[lane];
  op[lane + 32] = d1 * rstd * w[lane + 32] + bb[lane + 32];
  op[lane + 64] = d2 * rstd * w[lane + 64] + bb[lane + 64];
}

// ---------------- 2) in_proj: xz[25088,384] = xn[25088,96] x W^T ----------------
// One wave computes a 16x64 output strip (4 N-tiles share one A-fragment).
__global__ void k_gemm_inproj(const float* __restrict__ xn,
                              const float* __restrict__ w,  // [384,96]
                              float* __restrict__ xz) {
  int tile = (blockIdx.x * blockDim.x + threadIdx.x) >> 5;
  int lane = threadIdx.x & 31;
  constexpr int NB = OO_ / 64;  // 6 n-blocks
  int nb = tile % NB, mt = tile / NB;
  if (mt >= NP_ / 16) return;
  int m0 = mt * 16, n0 = nb * 64;
  int hl = lane >> 4, lr = lane & 15;
  const float* arow = xn + (long)(m0 + lr) * C_;
  const float* b0 = w + (long)(n0 +  0 + lr) * C_;  // B(kc,n) = W[n,kc]
  const float* b1 = w + (long)(n0 + 16 + lr) * C_;
  const float* b2 = w + (long)(n0 + 32 + lr) * C_;
  const float* b3 = w + (long)(n0 + 48 + lr) * C_;
  v8f ac0 = {}, ac1 = {}, ac2 = {}, ac3 = {};
  for (int kk = 0; kk < C_; kk += 4) {
    int kb = kk + 2 * hl;
    v2f a  = {arow[kb], arow[kb + 1]};
    v2f f0 = {b0[kb], b0[kb + 1]};
    v2f f1 = {b1[kb], b1[kb + 1]};
    v2f f2 = {b2[kb], b2[kb + 1]};
    v2f f3 = {b3[kb], b3[kb + 1]};
    ac0 = wmma4(a, f0, ac0);
    ac1 = wmma4(a, f1, ac1);
    ac2 = wmma4(a, f2, ac2);
    ac3 = wmma4(a, f3, ac3);
  }
  float* cp = xz + n0 + lr;
#pragma unroll
  for (int r = 0; r < 8; ++r) {
    long ro = (long)(m0 + r + 8 * hl) * OO_;
    cp[ro]      = ac0[r];
    cp[ro + 16] = ac1[r];
    cp[ro + 32] = ac2[r];
    cp[ro + 48] = ac3[r];
  }
}

// ---------------- 3) depthwise 3x3 conv + bias + SiLU ----------------
__global__ void k_conv(const float* __restrict__ xz, const float* __restrict__ cw,
                       const float* __restrict__ cb, float* __restrict__ xc) {
  int idx = blockIdx.x * blockDim.x + threadIdx.x;
  if (idx >= NP_ * D_) return;
  int d = idx % D_;
  int pg = idx / D_;  // b*L + p
  int p = pg % L_, b = pg / L_;
  int hh = p / W_, ww = p % W_;
  const float* w9 = cw + d * 9;
  float acc = cb[d];
#pragma unroll
  for (int ky = 0; ky < 3; ++ky) {
    int y = hh + ky - 1;
    if (y < 0 || y >= H_) continue;
#pragma unroll
    for (int kx = 0; kx < 3; ++kx) {
      int x = ww + kx - 1;
      if (x < 0 || x >= W_) continue;
      acc += xz[(long)(b * L_ + y * W_ + x) * OO_ + d] * w9[ky * 3 + kx];
    }
  }
  float sg = 1.f / (1.f + __expf(-acc));
  xc[(long)pg * D_ + d] = acc * sg;
}

// ---------------- 4) x_proj: xd[bk,t,38] = xs_k[t,:] x Wk^T (gathered rows) ----------------
// One wave computes all 3 padded N-tiles (cols 0..47, stores guarded to <38).
__global__ void k_gemm_xproj(const float* __restrict__ xc,
                             const float* __restrict__ w,  // [K,38,192]
                             float* __restrict__ xd) {
  int tile = (blockIdx.x * blockDim.x + threadIdx.x) >> 5;
  int lane = threadIdx.x & 31;
  constexpr int MT = L_ / 16;  // 196
  int mt = tile % MT;
  int bk = tile / MT;
  if (bk >= BK_) return;
  int k = bk & 3, b = bk >> 2;
  int m0 = mt * 16;
  int hl = lane >> 4, lr = lane & 15;
  int t = m0 + lr;
  int p = map_scan(t, k);
  const float* arow = xc + ((long)b * L_ + p) * D_;
  const float* wk = w + (long)k * CD_ * D_;
  int nc0 = lr, nc1 = 16 + lr, nc2 = 32 + lr;
  bool ok2 = (nc2 < CD_);
  const float* b0 = wk + (long)nc0 * D_;
  const float* b1 = wk + (long)nc1 * D_;
  const float* b2 = wk + (long)(ok2 ? nc2 : CD_ - 1) * D_;
  v8f ac0 = {}, ac1 = {}, ac2 = {};
  for (int kk = 0; kk < D_; kk += 4) {
    int kb = kk + 2 * hl;
    v2f a  = {arow[kb], arow[kb + 1]};
    v2f f0 = {b0[kb], b0[kb + 1]};
    v2f f1 = {b1[kb], b1[kb + 1]};
    v2f f2 = {ok2 ? b2[kb] : 0.f, ok2 ? b2[kb + 1] : 0.f};
    ac0 = wmma4(a, f0, ac0);
    ac1 = wmma4(a, f1, ac1);
    ac2 = wmma4(a, f2, ac2);
  }
  float* base = xd + (long)bk * L_ * CD_;
#pragma unroll
  for (int r = 0; r < 8; ++r) {
    long ro = (long)(m0 + r + 8 * hl) * CD_;
    base[ro + nc0] = ac0[r];
    base[ro + nc1] = ac1[r];
    if (ok2) base[ro + nc2] = ac2[r];
  }
}

// ---------------- 5/7) selective scan, chunked affine-scan decomposition ----------------
// PASS 1: from h0=0 compute per-chunk (P = prod dA, Q = h_end).
// PASS 2: from true chunk-entry state Hin, emit y + u*Ds.
template <int PASS>
__global__ void k_scan(const float* __restrict__ xc, const float* __restrict__ xd,
                       const float* __restrict__ alog, const float* __restrict__ dtw,
                       const float* __restrict__ dtb, const float* __restrict__ dss,
                       const float* __restrict__ Hin, float* __restrict__ Pb,
                       float* __restrict__ Qb, float* __restrict__ oy) {
  __shared__ float sbuf[CH_ * CD_];
  int bid = blockIdx.x;
  int s = bid % S_, bk = bid / S_;
  int k = bk & 3, b = bk >> 2;
  int d = threadIdx.x;  // 0..191
  int kd = k * D_ + d;
  float Ar[N_], hs[N_], Pp[N_], wr[R_];
#pragma unroll
  for (int n = 0; n < N_; ++n) Ar[n] = -__expf(alog[(long)kd * N_ + n]);
#pragma unroll
  for (int r = 0; r < R_; ++r) wr[r] = dtw[(long)kd * R_ + r];
  float bias = dtb[kd], Dd = dss[kd];
  if (PASS == 2) {
#pragma unroll
    for (int n = 0; n < N_; ++n)
      hs[n] = Hin[((long)(bk * S_ + s)) * (D_ * N_) + d * N_ + n];
  } else {
#pragma unroll
    for (int n = 0; n < N_; ++n) { hs[n] = 0.f; Pp[n] = 1.f; }
  }
  // stage x_dbl chunk (dt rank 0..5 | B 6..21 | C 22..37) into LDS
  const float* src = xd + ((long)bk * L_ + (long)s * CH_) * CD_;
  for (int i = d; i < CH_ * CD_; i += D_) sbuf[i] = src[i];
  __syncthreads();

  const float* xcb = xc + (long)b * L_ * D_ + d;
  for (int t = 0; t < CH_; ++t) {
    int tg = s * CH_ + t;
    int p = map_scan(tg, k);
    float u = xcb[(long)p * D_];
    const float* sv = sbuf + t * CD_;
    float acc = bias;
#pragma unroll
    for (int r = 0; r < R_; ++r) acc += wr[r] * sv[r];
    float dt = (acc > 20.f) ? acc : log1pf(__expf(acc));  // softplus
    float dtu = dt * u;
    float y = 0.f;
#pragma unroll
    for (int n = 0; n < N_; ++n) {
      float dA = __expf(dt * Ar[n]);
      hs[n] = hs[n] * dA + dtu * sv[R_ + n];
      if (PASS == 1) Pp[n] *= dA;
      else y += hs[n] * sv[R_ + N_ + n];
    }
    if (PASS == 2)
      oy[((long)bk * L_ + tg) * D_ + d] = y + u * Dd;
  }
  if (PASS == 1) {
    long o = ((long)(bk * S_ + s)) * (D_ * N_) + d * N_;
#pragma unroll
    for (int n = 0; n < N_; ++n) { Pb[o + n] = Pp[n]; Qb[o + n] = hs[n]; }
  }
}

// ---------------- 6) serial combine across chunks per (bk,d,n) ----------------
__global__ void k_combine(const float* __restrict__ Pb, const float* __restrict__ Qb,
                          float* __restrict__ Hin) {
  int idx = blockIdx.x * blockDim.x + threadIdx.x;
  if (idx >= BK_ * D_ * N_) return;
  int dn = idx % (D_ * N_);
  int bk = idx / (D_ * N_);
  float h = 0.f;
  for (int s = 0; s < S_; ++s) {
    long o = ((long)(bk * S_ + s)) * (D_ * N_) + dn;
    Hin[o] = h;
    h = Pb[o] * h + Qb[o];
  }
}

// ---------------- 8) merge 4 directions + out_norm + SiLU gate ----------------
__global__ void k_merge(const float* __restrict__ oy, const float* __restrict__ xz,
                        const float* __restrict__ nw, const float* __restrict__ nb,
                        float* __restrict__ yg) {
  int wave = (blockIdx.x * blockDim.x + threadIdx.x) >> 5;
  int lane = threadIdx.x & 31;
  if (wave >= NP_) return;
  int b = wave / L_, p = wave % L_;
  int hh = p / W_, ww = p % W_;
  int t1 = ww * H_ + hh;
  long base = (long)b * K_ * L_ * D_;
  const float* y0 = oy + base + (long)(0 * L_ + p) * D_;
  const float* y1 = oy + base + (long)(1 * L_ + t1) * D_;
  const float* y2 = oy + base + (long)(2 * L_ + (L_ - 1 - p)) * D_;
  const float* y3 = oy + base + (long)(3 * L_ + (L_ - 1 - t1)) * D_;
  float v[6], s = 0.f;
#pragma unroll
  for (int j = 0; j < 6; ++j) {
    int c = lane + 32 * j;
    v[j] = y0[c] + y1[c] + y2[c] + y3[c];
    s += v[j];
  }
  float mean = wave_sum(s) * (1.f / D_);
  float vs = 0.f;
#pragma unroll
  for (int j = 0; j < 6; ++j) { v[j] -= mean; vs += v[j] * v[j]; }
  float rstd = rsqrtf(wave_sum(vs) * (1.f / D_) + EPS_);
  const float* zp = xz + (long)wave * OO_ + D_;
  float* op = yg + (long)wave * D_;
#pragma unroll
  for (int j = 0; j < 6; ++j) {
    int c = lane + 32 * j;
    float z = zp[c];
    float sg = 1.f / (1.f + __expf(-z));
    op[c] = (v[j] * rstd * nw[c] + nb[c]) * (z * sg);
  }
}

// ---------------- 9) out_proj GEMM + residual ----------------
// One wave computes a full 16x96 output strip (6 N-tiles share one A-fragment).
__global__ void k_gemm_outproj(const float* __restrict__ yg,
                               const float* __restrict__ w,   // [96,192]
                               const float* __restrict__ xin, // residual
                               float* __restrict__ out) {
  int mt = (blockIdx.x * blockDim.x + threadIdx.x) >> 5;
  int lane = threadIdx.x & 31;
  if (mt >= NP_ / 16) return;
  int m0 = mt * 16;
  int hl = lane >> 4, lr = lane & 15;
  const float* arow = yg + (long)(m0 + lr) * D_;
  const float* bp[6];
#pragma unroll
  for (int j = 0; j < 6; ++j) bp[j] = w + (long)(j * 16 + lr) * D_;
  v8f ac[6] = {{}, {}, {}, {}, {}, {}};
  for (int kk = 0; kk < D_; kk += 4) {
    int kb = kk + 2 * hl;
    v2f a = {arow[kb], arow[kb + 1]};
#pragma unroll
    for (int j = 0; j < 6; ++j) {
      v2f f = {bp[j][kb], bp[j][kb + 1]};
      ac[j] = wmma4(a, f, ac[j]);
    }
  }
#pragma unroll
  for (int r = 0; r < 8; ++r) {
    long ro = (long)(m0 + r + 8 * hl) * C_ + lr;
#pragma unroll
    for (int j = 0; j < 6; ++j) {
      long o = ro + j * 16;
      out[o] = xin[o] + ac[j][r];
    }
  }
}

extern "C" void kernel_launch(void* const* d_in, const int* in_sizes, int n_in,
                              void* d_out, int out_size, void* d_ws, size_t ws_size,
                              hipStream_t stream) {
  (void)in_sizes; (void)n_in; (void)out_size; (void)ws_size;
  const float* x_in = (const float*)d_in[0];
  const float* ln1w = (const float*)d_in[1];
  const float* ln1b = (const float*)d_in[2];
  const float* wip  = (const float*)d_in[3];
  const float* cw   = (const float*)d_in[4];
  const float* cb   = (const float*)d_in[5];
  const float* wx   = (const float*)d_in[6];
  const float* wdt  = (const float*)d_in[7];
  const float* bdt  = (const float*)d_in[8];
  const float* alog = (const float*)d_in[9];
  const float* dss  = (const float*)d_in[10];
  const float* onw  = (const float*)d_in[11];
  const float* onb  = (const float*)d_in[12];
  const float* wop  = (const float*)d_in[13];
  float* out = (float*)d_out;

  float* ws = (float*)d_ws;
  size_t off = 0;
  float* xn = ws + off; off += (size_t)NP_ * C_;
  float* xz = ws + off; off += (size_t)NP_ * OO_;
  float* xc = ws + off; off += (size_t)NP_ * D_;
  float* xd = ws + off; off += (size_t)BK_ * L_ * CD_;
  float* Pb = ws + off; off += (size_t)BK_ * S_ * D_ * N_;
  float* Qb = ws + off; off += (size_t)BK_ * S_ * D_ * N_;
  float* Hi = ws + off; off += (size_t)BK_ * S_ * D_ * N_;
  float* oy = ws + off; off += (size_t)BK_ * L_ * D_;
  float* yg = ws + off;

  // 1) LayerNorm (one wave32 per pixel)
  k_ln1<<<NP_ / 8, 256, 0, stream>>>(x_in, ln1w, ln1b, xn);
  // 2) in_proj WMMA GEMM: 1568 m-tiles x 6 n-blocks (16x64 per wave)
  {
    int tiles = (NP_ / 16) * (OO_ / 64);
    k_gemm_inproj<<<(tiles + 3) / 4, 128, 0, stream>>>(xn, wip, xz);
  }
  // 3) depthwise conv + SiLU
  k_conv<<<(NP_ * D_ + 255) / 256, 256, 0, stream>>>(xz, cw, cb, xc);
  // 4) x_proj WMMA GEMM with direction-gathered rows: 32 x 196 strips (16x48 per wave)
  {
    int tiles = BK_ * (L_ / 16);
    k_gemm_xproj<<<(tiles + 3) / 4, 128, 0, stream>>>(xc, wx, xd);
  }
  // 5) scan pass 1 (per-chunk affine summary), 1568 blocks
  k_scan<1><<<BK_ * S_, D_, 0, stream>>>(xc, xd, alog, wdt, bdt, dss,
                                         nullptr, Pb, Qb, nullptr);
  // 6) combine chunk states serially (tiny)
  k_combine<<<(BK_ * D_ * N_ + 255) / 256, 256, 0, stream>>>(Pb, Qb, Hi);
  // 7) scan pass 2 (emit y), 1568 blocks
  k_scan<2><<<BK_ * S_, D_, 0, stream>>>(xc, xd, alog, wdt, bdt, dss,
                                         Hi, nullptr, nullptr, oy);
  // 8) merge 4 directions + out_norm + gate
  k_merge<<<NP_ / 8, 256, 0, stream>>>(oy, xz, onw, onb, yg);
  // 9) out_proj WMMA GEMM + residual: 1568 strips (16x96 per wave)
  k_gemm_outproj<<<((NP_ / 16) + 3) / 4, 128, 0, stream>>>(yg, wop, x_in, out);
}